// SSM_spa_40458591928874
// MI455X (gfx1250) — compile-verified
//
#include <hip/hip_runtime.h>
#include <hip/hip_bf16.h>
#include <math.h>

// ---------------------------------------------------------------------------
// Mamba-style SSM block for MI455X (gfx1250, wave32).
// All GEMMs run on V_WMMA_F32_16X16X4_F32 (native fp32 WMMA) for exact
// precision vs. the fp32 reference.  One wave computes a TM*16 x TN*16 tile
// (register-blocked: A fragments reused across TN, B fragments across TM).
// ---------------------------------------------------------------------------

typedef __attribute__((ext_vector_type(2))) float v2f;
typedef __attribute__((ext_vector_type(8))) float v8f;

#define D_MODEL 256
#define D_INNER 512
#define D_STATE 16
#define DT_RANK 16
#define NBATCH  8
#define SEQ     512              // = c of the input image
#define MROWS   (NBATCH * SEQ)   // 4096

// ---------------------------------------------------------------------------
// 1) bilinear 64x64 -> 16x16  (linspace weights: fy = oy*63/15)
//    output X[m][256], m = b*512 + ch, col = oy*16+ox
// ---------------------------------------------------------------------------
__global__ void k_downsample(const float* __restrict__ in, float* __restrict__ X) {
  int idx = blockIdx.x * blockDim.x + threadIdx.x;      // 8*512*256 = 1,048,576
  if (idx >= MROWS * D_MODEL) return;
  int o = idx & 255;
  int m = idx >> 8;
  int oy = o >> 4, ox = o & 15;
  float fy = oy * (63.0f / 15.0f);
  float fx = ox * (63.0f / 15.0f);
  int y0 = min((int)floorf(fy), 63); int y1 = min(y0 + 1, 63); float wy = fy - (float)y0;
  int x0 = min((int)floorf(fx), 63); int x1 = min(x0 + 1, 63); float wx = fx - (float)x0;
  const float* p = in + (size_t)m * 4096;
  float v00 = p[y0 * 64 + x0], v10 = p[y1 * 64 + x0];
  float v01 = p[y0 * 64 + x1], v11 = p[y1 * 64 + x1];
  float r0 = v00 + (v10 - v00) * wy;
  float r1 = v01 + (v11 - v01) * wy;
  X[idx] = r0 + (r1 - r0) * wx;
}

// ---------------------------------------------------------------------------
// 2) WMMA fp32 GEMM:  C[M x N] = A[M x K(lda)] * W[N x K(ldw)]^T
//    Register-blocked TM x TN sub-tiles of 16x16 per wave.
//    EPI==1: bias + softplus epilogue (dt projection).
//    Grid must cover exactly (M/(16*TM))*(N/(16*TN)) waves.
// ---------------------------------------------------------------------------
template <int TM, int TN, int EPI>
__global__ void k_wmma_gemm(const float* __restrict__ A, int lda,
                            const float* __restrict__ W, int ldw,
                            float* __restrict__ C, int ldc,
                            const float* __restrict__ bias,
                            int M, int N, int K) {
  int wave = (blockIdx.x * blockDim.x + threadIdx.x) >> 5;
  int lane = threadIdx.x & 31;
  int tiles_n = N / (16 * TN);
  int tm = wave / tiles_n;
  int tn = wave - tm * tiles_n;
  if (tm * 16 * TM >= M) return;            // wave-uniform; EXEC stays all-ones

  int row = lane & 15;                      // M-row for A frag / N-col for B frag
  int hi  = lane >> 4;                      // selects K pair {0,1} vs {2,3}
  const float* Arow = A + (size_t)(tm * 16 * TM + row) * lda + 2 * hi;
  const float* Wrow = W + (size_t)(tn * 16 * TN + row) * ldw + 2 * hi;

  v8f acc[TM][TN];
#pragma unroll
  for (int i = 0; i < TM; ++i)
#pragma unroll
    for (int t = 0; t < TN; ++t) acc[i][t] = (v8f){};

  for (int k = 0; k < K; k += 4) {
    v2f a[TM], b[TN];
#pragma unroll
    for (int i = 0; i < TM; ++i)
      a[i] = *(const v2f*)(Arow + (size_t)i * 16 * lda + k);
#pragma unroll
    for (int t = 0; t < TN; ++t)
      b[t] = *(const v2f*)(Wrow + (size_t)t * 16 * ldw + k);
#pragma unroll
    for (int i = 0; i < TM; ++i)
#pragma unroll
      for (int t = 0; t < TN; ++t)
        acc[i][t] = __builtin_amdgcn_wmma_f32_16x16x4_f32(
            /*neg_a=*/false, a[i], /*neg_b=*/false, b[t],
            /*c_mod=*/(short)0, acc[i][t], /*reuse_a=*/false, /*reuse_b=*/false);
  }

#pragma unroll
  for (int i = 0; i < TM; ++i) {
#pragma unroll
    for (int t = 0; t < TN; ++t) {
      int col = (tn * TN + t) * 16 + row;
      float bv = (EPI == 1) ? bias[col] : 0.0f;
#pragma unroll
      for (int r = 0; r < 8; ++r) {
        int mrow = (tm * TM + i) * 16 + r + 8 * hi;  // C/D: VGPR r -> M = r + 8*hi
        float v = acc[i][t][r];
        if (EPI == 1) {                              // softplus(v + bias)
          v += bv;
          v = (v > 20.0f) ? v : log1pf(expf(v));
        }
        C[(size_t)mrow * ldc + col] = v;
      }
    }
  }
}

// ---------------------------------------------------------------------------
// 3) depthwise causal conv (k=4, left pad 3) + SiLU:
//    u[m][d] = silu( sum_j w[d][j]*xp[b][l+j-3][d] + cb[d] ),  xp = XZ[:, :512]
// ---------------------------------------------------------------------------
__global__ void k_conv_silu(const float* __restrict__ XZ,
                            const float* __restrict__ cw,
                            const float* __restrict__ cb,
                            float* __restrict__ U) {
  int idx = blockIdx.x * blockDim.x + threadIdx.x;      // 8*512*512
  if (idx >= MROWS * D_INNER) return;
  int d = idx & (D_INNER - 1);
  int m = idx >> 9;
  int l = m & (SEQ - 1);
  int b = m >> 9;
  float acc = cb[d];
  const float* w = cw + d * 4;
#pragma unroll
  for (int j = 0; j < 4; ++j) {
    int ll = l + j - 3;
    if (ll >= 0)
      acc += w[j] * XZ[(size_t)(b * SEQ + ll) * (2 * D_INNER) + d];
  }
  U[idx] = acc / (1.0f + expf(-acc));                   // silu
}

// ---------------------------------------------------------------------------
// 4) selective scan + gating, fused.  One block per batch, thread d owns the
//    16-state recurrence for channel d; B/C broadcast through LDS per step.
//    Writes YG[m][d] = (y + u*D[d]) * silu(z)
// ---------------------------------------------------------------------------
__global__ void __launch_bounds__(D_INNER)
k_scan_gate(const float* __restrict__ DT, const float* __restrict__ XDBL,
            const float* __restrict__ U,  const float* __restrict__ XZ,
            const float* __restrict__ A_log, const float* __restrict__ Dp,
            float* __restrict__ YG) {
  int b = blockIdx.x;
  int d = threadIdx.x;
  __shared__ float sBC[2 * D_STATE];

  float Ar[D_STATE];
#pragma unroll
  for (int n = 0; n < D_STATE; ++n) Ar[n] = -expf(A_log[d * D_STATE + n]);
  float h[D_STATE];
#pragma unroll
  for (int n = 0; n < D_STATE; ++n) h[n] = 0.0f;
  float Dd = Dp[d];

  for (int l = 0; l < SEQ; ++l) {
    int m = b * SEQ + l;
    if (d < 2 * D_STATE)                    // x_dbl cols 16..47 = [B | C]
      sBC[d] = XDBL[(size_t)m * (DT_RANK + 2 * D_STATE) + DT_RANK + d];
    __syncthreads();

    float dt = DT[(size_t)m * D_INNER + d];
    float u  = U[(size_t)m * D_INNER + d];
    float y = 0.0f;
#pragma unroll
    for (int n = 0; n < D_STATE; ++n) {
      float dA = expf(dt * Ar[n]);
      h[n] = dA * h[n] + dt * sBC[n] * u;
      y += h[n] * sBC[D_STATE + n];
    }
    float z  = XZ[(size_t)m * (2 * D_INNER) + D_INNER + d];
    float sz = z / (1.0f + expf(-z));
    YG[(size_t)m * D_INNER + d] = (y + u * Dd) * sz;
    __syncthreads();
  }
}

// ---------------------------------------------------------------------------
// 5) bilinear 16x16 -> 64x64 (fy = oy*15/63); S[m][256] -> out (8,512,64,64)
// ---------------------------------------------------------------------------
__global__ void k_upsample(const float* __restrict__ S, float* __restrict__ out) {
  int idx = blockIdx.x * blockDim.x + threadIdx.x;      // 8*512*4096 = 16,777,216
  if (idx >= MROWS * 4096) return;
  int pix = idx & 4095;
  int m = idx >> 12;
  int oy = pix >> 6, ox = pix & 63;
  float fy = oy * (15.0f / 63.0f);
  float fx = ox * (15.0f / 63.0f);
  int y0 = min((int)floorf(fy), 15); int y1 = min(y0 + 1, 15); float wy = fy - (float)y0;
  int x0 = min((int)floorf(fx), 15); int x1 = min(x0 + 1, 15); float wx = fx - (float)x0;
  const float* p = S + (size_t)m * 256;
  float v00 = p[y0 * 16 + x0], v10 = p[y1 * 16 + x0];
  float v01 = p[y0 * 16 + x1], v11 = p[y1 * 16 + x1];
  float r0 = v00 + (v10 - v00) * wy;
  float r1 = v01 + (v11 - v01) * wy;
  out[idx] = r0 + (r1 - r0) * wx;
}

// ---------------------------------------------------------------------------
// launcher
// ---------------------------------------------------------------------------
extern "C" void kernel_launch(void* const* d_in, const int* in_sizes, int n_in,
                              void* d_out, int out_size, void* d_ws, size_t ws_size,
                              hipStream_t stream) {
  const float* x1        = (const float*)d_in[0];
  const float* in_proj_w = (const float*)d_in[1];   // (1024, 256)
  const float* conv_w    = (const float*)d_in[2];   // (512, 1, 4)
  const float* conv_b    = (const float*)d_in[3];   // (512,)
  const float* x_proj_w  = (const float*)d_in[4];   // (48, 512)
  const float* dt_proj_w = (const float*)d_in[5];   // (512, 16)
  const float* dt_proj_b = (const float*)d_in[6];   // (512,)
  const float* A_log     = (const float*)d_in[7];   // (512, 16)
  const float* Dp        = (const float*)d_in[8];   // (512,)
  const float* out_proj_w= (const float*)d_in[9];   // (256, 512)
  float* out = (float*)d_out;

  // workspace layout (floats)
  float* ws   = (float*)d_ws;
  float* X    = ws;                                  // 4096*256
  float* XZ   = X    + (size_t)MROWS * D_MODEL;      // 4096*1024
  float* U    = XZ   + (size_t)MROWS * 2 * D_INNER;  // 4096*512
  float* XDBL = U    + (size_t)MROWS * D_INNER;      // 4096*48
  float* DT   = XDBL + (size_t)MROWS * (DT_RANK + 2 * D_STATE); // 4096*512
  float* YG   = DT   + (size_t)MROWS * D_INNER;      // 4096*512
  float* OUTS = YG   + (size_t)MROWS * D_INNER;      // 4096*256

  // 1) downsample
  {
    int n = MROWS * D_MODEL;
    k_downsample<<<(n + 255) / 256, 256, 0, stream>>>(x1, X);
  }
  // 2) in_proj: XZ = X(4096x256) * in_proj_w(1024x256)^T   (TM=2, TN=4)
  {
    int waves = (MROWS / 32) * (1024 / 64);   // 128*16 = 2048 -> 256 blocks
    k_wmma_gemm<2, 4, 0><<<waves / 8, 256, 0, stream>>>(
        X, D_MODEL, in_proj_w, D_MODEL, XZ, 2 * D_INNER, nullptr,
        MROWS, 2 * D_INNER, D_MODEL);
  }
  // 3) depthwise conv + silu -> U
  {
    int n = MROWS * D_INNER;
    k_conv_silu<<<(n + 255) / 256, 256, 0, stream>>>(XZ, conv_w, conv_b, U);
  }
  // 4) x_proj: XDBL = U(4096x512) * x_proj_w(48x512)^T   (TM=2, TN=3: full 48)
  {
    int waves = (MROWS / 32) * (48 / 48);     // 128 -> 16 blocks
    k_wmma_gemm<2, 3, 0><<<waves / 8, 256, 0, stream>>>(
        U, D_INNER, x_proj_w, D_INNER, XDBL, DT_RANK + 2 * D_STATE, nullptr,
        MROWS, 48, D_INNER);
  }
  // 5) dt: DT = softplus( XDBL[:, :16] * dt_proj_w(512x16)^T + b )   (TM=2, TN=4)
  {
    int waves = (MROWS / 32) * (D_INNER / 64); // 128*8 = 1024 -> 128 blocks
    k_wmma_gemm<2, 4, 1><<<waves / 8, 256, 0, stream>>>(
        XDBL, DT_RANK + 2 * D_STATE, dt_proj_w, DT_RANK, DT, D_INNER, dt_proj_b,
        MROWS, D_INNER, DT_RANK);
  }
  // 6) selective scan + gating -> YG
  k_scan_gate<<<NBATCH, D_INNER, 0, stream>>>(DT, XDBL, U, XZ, A_log, Dp, YG);
  // 7) out_proj: OUTS = YG(4096x512) * out_proj_w(256x512)^T   (TM=2, TN=4)
  {
    int waves = (MROWS / 32) * (256 / 64);    // 128*4 = 512 -> 64 blocks
    k_wmma_gemm<2, 4, 0><<<waves / 8, 256, 0, stream>>>(
        YG, D_INNER, out_proj_w, D_INNER, OUTS, 256, nullptr,
        MROWS, 256, D_INNER);
  }
  // 8) upsample -> d_out (8,512,64,64)
  {
    int n = MROWS * 4096;
    k_upsample<<<(n + 255) / 256, 256, 0, stream>>>(OUTS, out);
  }
  (void)in_sizes; (void)n_in; (void)out_size; (void)ws_size;
}